// AleatoricClassificationLoss_62732292325851
// MI455X (gfx1250) — compile-verified
//
#include <hip/hip_runtime.h>
#include <cstdint>

// Aleatoric classification loss, B=1024, C=1000, S=100.
// Transcendental/VALU bound (102M softmax exps + 102M Gaussian draws), not
// bandwidth bound (8 MB inputs vs 23.3 TB/s). Strategy: stage exp(log_std) in
// LDS once per block, hold the per-wave working set (logits, std, prd row) in
// registers so the hot loops are pure VALU/TRANS/WMMA with no memory waits.
// V_WMMA_F32_16X16X4_F32 (full fp32) accumulates the softmax denominator
// 64 values/instruction on the matrix pipe, co-executing with the exp stream.
// Box-Muller is paired (cos+sin) so two Gaussians share one hash+log+sqrt.

#define NCLS   1000
#define NPAD   1024
#define NSAMP  100
#define NWAVES 8
#define NCHUNK 32            // classes per lane (NPAD / 32 lanes)

typedef __attribute__((ext_vector_type(2))) float v2f;
typedef __attribute__((ext_vector_type(8))) float v8f;

__device__ __forceinline__ uint32_t hash32(uint32_t x) {
    x ^= x >> 16; x *= 0x7feb352dU;
    x ^= x >> 15; x *= 0x846ca68bU;
    x ^= x >> 16;
    return x;
}

// Deterministic counter-based pair of N(0,1) draws: one Box-Muller transform
// yields (r*cos, r*sin), sharing the hash/convert/log2/sqrt across two values.
// ln(u) folded through the native log2: -2*ln(u) = -2*ln2*log2(u).
__device__ __forceinline__ v2f gauss_pair(uint32_t pair_idx) {
    uint32_t h1 = hash32(pair_idx * 2u + 0x9e3779b9u);
    uint32_t h2 = hash32(pair_idx * 2u + 0x85ebca6bu);
    float u1 = (float)(h1 >> 8) * (1.0f / 16777216.0f);
    float u2 = (float)(h2 >> 8) * (1.0f / 16777216.0f);
    u1 = fmaxf(u1, 1.0f / 16777216.0f);                  // avoid log(0)
    float r  = __builtin_sqrtf(-1.38629436111989f * __log2f(u1));  // -2*ln2
    float th = 6.28318530717958647692f * u2;
    v2f z;
    z[0] = r * __cosf(th);
    z[1] = r * __sinf(th);
    return z;
}

__global__ __launch_bounds__(256)
void aleatoric_loss_main(const float* __restrict__ logits,
                         const int*   __restrict__ targets,
                         const float* __restrict__ log_std,
                         float*       __restrict__ partials) {
    __shared__ float s_logits[NPAD];
    __shared__ float s_std[NPAD];
    __shared__ float s_part[NWAVES];

    const int b   = blockIdx.x;
    const int tid = threadIdx.x;

    // Stage row b once per block: logits and exp(log_std) (std-exp computed
    // once per block, not once per wave). Pad 1000 -> 1024.
    for (int i = tid; i < NPAD; i += 256) {
        bool v = (i < NCLS);
        s_logits[i] = v ? logits[b * NCLS + i] : 0.0f;
        s_std[i]    = v ? __expf(log_std[b * NCLS + i]) : 0.0f;
    }
    const int t = targets[b];
    __syncthreads();

    const int wave = tid >> 5;
    const int lane = tid & 31;

    // Pull the row into registers once per wave: lane owns c = lane + 32*j.
    float rlog[NCHUNK], rstd[NCHUNK];
    #pragma unroll
    for (int j = 0; j < NCHUNK; ++j) {
        rlog[j] = s_logits[lane + 32 * j];
        rstd[j] = s_std[lane + 32 * j];
    }

    const v2f ones = {1.0f, 1.0f};  // B-matrix (4x16) of ones: WMMA sums A
    float nll_acc = 0.0f;

    for (int s = wave; s < NSAMP; s += NWAVES) {  // wave-uniform -> EXEC full
        // ---- Pass 1: paired Gaussian draws, prd row in registers, running max.
        float prd[NCHUNK];
        float m     = -3.402823466e38f;
        float prd_t = 0.0f;
        #pragma unroll
        for (int i = 0; i < NCHUNK / 2; ++i) {
            const int j0 = 2 * i, j1 = 2 * i + 1;
            const int c0 = lane + 32 * j0;
            const int c1 = lane + 32 * j1;
            uint32_t pidx = ((uint32_t)b * (uint32_t)NCLS + (uint32_t)c0)
                            * (uint32_t)NSAMP + (uint32_t)s;
            v2f z = gauss_pair(pidx);
            float p0 = __fmaf_rn(z[0], rstd[j0], rlog[j0]);
            float p1 = __fmaf_rn(z[1], rstd[j1], rlog[j1]);
            if (j1 == NCHUNK - 1)                  // compile-time except last
                p1 = (c1 < NCLS) ? p1 : -__builtin_inff(); // exp(-inf-m)==0
            prd[j0] = p0;
            prd[j1] = p1;
            m = fmaxf(m, fmaxf(p0, p1));
            prd_t = (c0 == t) ? p0 : prd_t;
            prd_t = (c1 == t) ? p1 : prd_t;
        }
        #pragma unroll
        for (int off = 16; off > 0; off >>= 1)    // wave32 max-reduce
            m = fmaxf(m, __shfl_xor(m, off, 32));
        prd_t = __shfl(prd_t, t & 31, 32);        // broadcast target logit

        // ---- Pass 2: Sum_c exp(prd - m) via WMMA accumulate (A x ones + C).
        // Pure registers: exp stream on TRANS co-executes with the matrix pipe.
        v8f acc = {};
        #pragma unroll
        for (int i = 0; i < NCHUNK / 2; ++i) {
            v2f a;
            a[0] = __expf(prd[2 * i]     - m);
            a[1] = __expf(prd[2 * i + 1] - m);
            acc = __builtin_amdgcn_wmma_f32_16x16x4_f32(
                      false, a, false, ones, (short)0, acc, false, false);
        }
        // Lanes 0-15 hold rows 0-7 in acc[0..7]; lanes 16-31 hold rows 8-15.
        float ssum = acc[0] + acc[1] + acc[2] + acc[3]
                   + acc[4] + acc[5] + acc[6] + acc[7];
        ssum += __shfl_xor(ssum, 16, 32);

        // -log_softmax at target; ln via native log2.
        nll_acc += m + 0.69314718055994531f * __log2f(ssum) - prd_t;
    }

    if (lane == 0) s_part[wave] = nll_acc;        // nll_acc is wave-uniform
    __syncthreads();
    if (tid == 0) {
        float sum = 0.0f;
        #pragma unroll
        for (int w = 0; w < NWAVES; ++w) sum += s_part[w];
        partials[b] = sum;
    }
}

__global__ __launch_bounds__(256)
void aleatoric_loss_finalize(const float* __restrict__ partials,
                             float* __restrict__ out, int B) {
    __shared__ float s[256];
    float v = 0.0f;
    for (int i = threadIdx.x; i < B; i += 256) v += partials[i];
    s[threadIdx.x] = v;
    __syncthreads();
    for (int off = 128; off > 0; off >>= 1) {
        if (threadIdx.x < (unsigned)off) s[threadIdx.x] += s[threadIdx.x + off];
        __syncthreads();
    }
    if (threadIdx.x == 0) out[0] = s[0] / ((float)B * (float)NSAMP);
}

extern "C" void kernel_launch(void* const* d_in, const int* in_sizes, int n_in,
                              void* d_out, int out_size, void* d_ws, size_t ws_size,
                              hipStream_t stream) {
    const float* logits  = (const float*)d_in[0];
    const int*   targets = (const int*)  d_in[1];
    const float* log_std = (const float*)d_in[2];
    float*       out     = (float*)d_out;
    float*       parts   = (float*)d_ws;           // B floats of scratch

    const int B = in_sizes[1];                     // 1024

    aleatoric_loss_main<<<B, 256, 0, stream>>>(logits, targets, log_std, parts);
    aleatoric_loss_finalize<<<1, 256, 0, stream>>>(parts, out, B);
}